// HyperspectralAttention_34428457845108
// MI455X (gfx1250) — compile-verified
//
#include <hip/hip_runtime.h>

// CDNA5 WMMA fragment types (wave32):
//  A 16x4 f32  -> 2 VGPRs/lane, B 4x16 f32 -> 2 VGPRs/lane, C/D 16x16 f32 -> 8 VGPRs/lane
typedef __attribute__((ext_vector_type(2))) float v2f;
typedef __attribute__((ext_vector_type(8))) float v8f;

#define CH 64
#define HW 65536                 // 256*256 pixels per (batch, channel) plane
#define NBATCH 4
#define PX_PER_WAVE 16
#define WAVES_PER_WG 8
#define PX_PER_WG (PX_PER_WAVE * WAVES_PER_WG)   // 128
#define WG_PER_BATCH (HW / PX_PER_WG)            // 512
#define W_STRIDE 68              // padded LDS row stride for 64x64 weights (bank-conflict-free)
#define T_STRIDE 20              // padded LDS row stride for per-wave 64x16 T tile

__global__ __launch_bounds__(256)
void hyperspec_attn_fused(const float* __restrict__ x,
                          const float* __restrict__ illu,
                          const float* __restrict__ w_v_spec,
                          const float* __restrict__ w_v_spat,
                          const float* __restrict__ w_proj,
                          float* __restrict__ out)
{
    __shared__ float lds_wmix [CH * W_STRIDE];                // 0.6*Wv_spec + 0.4*Wv_spat
    __shared__ float lds_wproj[CH * W_STRIDE];
    __shared__ float lds_T[WAVES_PER_WG][CH * T_STRIDE];      // per-wave 64ch x 16px staging

    const int tid = threadIdx.x;

    // Cooperative weight preload + fold of the 0.6/0.4 mix (done once, amortized over 128 px)
    for (int i = tid; i < CH * CH; i += 256) {
        const int o = i >> 6, c = i & 63;
        lds_wmix [o * W_STRIDE + c] = 0.6f * w_v_spec[i] + 0.4f * w_v_spat[i];
        lds_wproj[o * W_STRIDE + c] = w_proj[i];
    }
    __syncthreads();

    const int wave  = tid >> 5;
    const int lane  = tid & 31;
    const int lm    = lane & 15;            // M (A) / N (B,C,D) index within a 16-wide tile
    const int khalf = (lane >> 4) << 1;     // lanes 16..31 carry K+2 / M+8 per ISA layout
    const int chadd = (lane >> 4) << 3;     // C/D: VGPR r holds M=r (lo lanes) / r+8 (hi lanes)

    const int wg    = blockIdx.x;
    const int batch = wg / WG_PER_BATCH;
    const int tile  = wg % WG_PER_BATCH;
    const int px    = tile * PX_PER_WG + wave * PX_PER_WAVE + lm;

    const size_t base = (size_t)batch * CH * HW;
    const float* __restrict__ xb = x    + base;
    const float* __restrict__ ib = illu + base;
    float*       __restrict__ ob = out  + base;

    // Warm L2/WGP$ for the illu rows we'll touch after GEMM1 (global_prefetch_b8)
    __builtin_prefetch(&ib[(size_t)lane        * HW + px], 0, 1);
    __builtin_prefetch(&ib[(size_t)(lane + 32) * HW + px], 0, 1);

    // ---------------- GEMM1: T(64 x 16px) = W_mix(64x64) * X(64x16px) ----------------
    v8f acc[4] = {};
    #pragma unroll
    for (int kk = 0; kk < 16; ++kk) {
        const int k0 = kk * 4 + khalf;
        v2f bfrag;                                  // B 4x16: lane=pixel, halves carry K,K+2
        bfrag.x = xb[(size_t)(k0    ) * HW + px];   // contiguous 64B per lane-half
        bfrag.y = xb[(size_t)(k0 + 1) * HW + px];
        #pragma unroll
        for (int m = 0; m < 4; ++m) {
            const float* wrow = &lds_wmix[(m * 16 + lm) * W_STRIDE + k0];
            v2f afrag;                              // A 16x4: lane=M, halves carry K,K+2
            afrag.x = wrow[0];
            afrag.y = wrow[1];
            acc[m] = __builtin_amdgcn_wmma_f32_16x16x4_f32(
                false, afrag, false, bfrag, (short)0, acc[m], false, false);
        }
    }

    // ---------------- elementwise illu, stage T through per-wave LDS ----------------
    #pragma unroll
    for (int m = 0; m < 4; ++m) {
        #pragma unroll
        for (int r = 0; r < 8; ++r) {
            const int ch = m * 16 + r + chadd;
            const float v = acc[m][r] * ib[(size_t)ch * HW + px];
            lds_T[wave][ch * T_STRIDE + lm] = v;    // wave-private; DS kept in order per wave
        }
    }

    // ---------------- GEMM2: OUT(64 x 16px) = W_proj(64x64) * T(64x16px) ----------------
    v8f acc2[4] = {};
    #pragma unroll
    for (int kk = 0; kk < 16; ++kk) {
        const int k0 = kk * 4 + khalf;
        v2f bfrag;
        bfrag.x = lds_T[wave][(k0    ) * T_STRIDE + lm];
        bfrag.y = lds_T[wave][(k0 + 1) * T_STRIDE + lm];
        #pragma unroll
        for (int m = 0; m < 4; ++m) {
            const float* wrow = &lds_wproj[(m * 16 + lm) * W_STRIDE + k0];
            v2f afrag;
            afrag.x = wrow[0];
            afrag.y = wrow[1];
            acc2[m] = __builtin_amdgcn_wmma_f32_16x16x4_f32(
                false, afrag, false, bfrag, (short)0, acc2[m], false, false);
        }
    }

    // ---------------- coalesced store: lane=pixel within each channel row ----------------
    #pragma unroll
    for (int m = 0; m < 4; ++m) {
        #pragma unroll
        for (int r = 0; r < 8; ++r) {
            const int ch = m * 16 + r + chadd;
            ob[(size_t)ch * HW + px] = acc2[m][r];
        }
    }
}

extern "C" void kernel_launch(void* const* d_in, const int* in_sizes, int n_in,
                              void* d_out, int out_size, void* d_ws, size_t ws_size,
                              hipStream_t stream) {
    // setup_inputs() order: 0:x 1:illu_feat 2:w_q_spec 3:w_k_spec 4:w_v_spec 5:w_q_spat
    //                       6:w_k_spat 7:w_v_spat 8:w_anchor_dw 9:w_anchor_pw 10:w_proj
    //                       11:temp_spec 12:temp_spat
    // All attention matrices are 1x1 per head -> softmax == 1 exactly; only
    // x, illu, w_v_spec, w_v_spat, w_proj are live.
    const float* x      = (const float*)d_in[0];
    const float* illu   = (const float*)d_in[1];
    const float* wvspec = (const float*)d_in[4];
    const float* wvspat = (const float*)d_in[7];
    const float* wproj  = (const float*)d_in[10];
    float* out = (float*)d_out;

    dim3 grid(NBATCH * WG_PER_BATCH);   // 2048 workgroups
    dim3 block(256);                    // 8 wave32s
    hipLaunchKernelGGL(hyperspec_attn_fused, grid, block, 0, stream,
                       x, illu, wvspec, wvspat, wproj, out);
}